// PEncoder_51711406243954
// MI455X (gfx1250) — compile-verified
//
#include <hip/hip_runtime.h>

// PEncoder for MI455X (gfx1250, wave32).
// Store-bandwidth-bound: 178.3 MB of f32 output -> ~7.7us floor @ 23.3 TB/s.
// No contraction dimension exists, so WMMA is inapplicable; optimization =
// coalesced b128 stores (RT hint: output fits the 192MB L2, wr-rinse drains),
// wave32 shuffle reduction, and CDNA5 async global->LDS staging of the input.

typedef float v4f __attribute__((ext_vector_type(4)));

#define N_ELEM 262144   // 1024 * 256
#define N_V4   65536    // N_ELEM / 4
#define M_POP  10
#define STEPS  16

// Monotonic float <-> uint key (works for any sign) for atomicMin/Max.
__device__ __forceinline__ unsigned f2key(float f) {
    unsigned b = __float_as_uint(f);
    return (b & 0x80000000u) ? ~b : (b | 0x80000000u);
}
__device__ __forceinline__ float key2f(unsigned k) {
    unsigned b = (k & 0x80000000u) ? (k ^ 0x80000000u) : ~k;
    return __uint_as_float(b);
}

__global__ void pe_init_ws(unsigned* __restrict__ ws) {
    ws[0] = 0xFFFFFFFFu;  // running min key
    ws[1] = 0x00000000u;  // running max key
}

// Global min/max: one float4 per thread, wave32 shuffle tree, 2 atomics/wave.
__global__ __launch_bounds__(256) void pe_minmax(const float* __restrict__ in,
                                                 unsigned* __restrict__ ws) {
    const int tid = blockIdx.x * blockDim.x + threadIdx.x;  // 0..N_V4-1
    v4f x = ((const v4f*)in)[tid];
    float mn = fminf(fminf(x.x, x.y), fminf(x.z, x.w));
    float mx = fmaxf(fmaxf(x.x, x.y), fmaxf(x.z, x.w));
#pragma unroll
    for (int off = 16; off > 0; off >>= 1) {  // wave32: 5 steps, width 32
        mn = fminf(mn, __shfl_xor(mn, off, 32));
        mx = fmaxf(mx, __shfl_xor(mx, off, 32));
    }
    if ((threadIdx.x & 31) == 0) {
        atomicMin(&ws[0], f2key(mn));
        atomicMax(&ws[1], f2key(mx));
    }
}

// Fused RBF drive + 16-step integrate-and-fire + rate. One float4 group per
// thread; 170 coalesced b128 stores per thread (512B/wave/instruction).
__global__ __launch_bounds__(256) void pe_encode(const float* __restrict__ in,
                                                 const float* __restrict__ vthp,
                                                 const unsigned* __restrict__ ws,
                                                 float* __restrict__ out) {
    const int tid = blockIdx.x * 256 + threadIdx.x;  // float4-group index

    // ---- CDNA5 async copy: stage this block's 4KB input tile into LDS ----
    // (single __shared__ object in this kernel -> LDS offset 0)
    __shared__ v4f tile[256];
    {
        unsigned lds_off = (unsigned)(threadIdx.x * sizeof(v4f));
        unsigned voff    = (unsigned)((unsigned)tid * sizeof(v4f));
        asm volatile(
            "global_load_async_to_lds_b128 %0, %1, %2\n\t"
            "s_wait_asynccnt 0"
            :
            : "v"(lds_off), "v"(voff), "s"(in), "v"(&tile[0])
            : "memory");
    }
    v4f x = tile[threadIdx.x];  // only this lane's slot is read: no barrier

    const float VTH   = vthp[0];
    const float imin  = key2f(ws[0]);
    const float imax  = key2f(ws[1]);
    const float span  = (imax - imin) * (1.0f / (float)(M_POP - 2));
    const float sigma = span * (1.0f / 1.5f);
    const float den   = 2.0f * sigma * sigma;

    v4f* o = (v4f*)out;
    const size_t plane = (size_t)N_V4;  // one (k,m) plane, in float4 units

    for (int m = 0; m < M_POP; ++m) {
        const float mu = imin + (2.0f * (float)m - 3.0f) * 0.5f * span;
        v4f diff = x - mu;
        v4f d;
        d.x = expf(-(diff.x * diff.x) / den);
        d.y = expf(-(diff.y * diff.y) / den);
        d.z = expf(-(diff.z * diff.z) / den);
        d.w = expf(-(diff.w * diff.w) / den);

        v4f v    = {0.0f, 0.0f, 0.0f, 0.0f};
        v4f rate = {0.0f, 0.0f, 0.0f, 0.0f};
        const size_t base = (size_t)m * plane + (size_t)tid;
#pragma unroll
        for (int k = 0; k < STEPS; ++k) {
            v += d;
            v4f s;
            s.x = (v.x >= VTH) ? 1.0f : 0.0f;
            s.y = (v.y >= VTH) ? 1.0f : 0.0f;
            s.z = (v.z >= VTH) ? 1.0f : 0.0f;
            s.w = (v.w >= VTH) ? 1.0f : 0.0f;
            v -= s * VTH;
            rate += s;
            o[(size_t)k * ((size_t)M_POP * plane) + base] = s;  // RT store: fits L2
        }
        o[(size_t)(STEPS * M_POP) * plane + base] = rate * (1.0f / (float)STEPS);
    }
}

extern "C" void kernel_launch(void* const* d_in, const int* in_sizes, int n_in,
                              void* d_out, int out_size, void* d_ws, size_t ws_size,
                              hipStream_t stream) {
    (void)in_sizes; (void)n_in; (void)out_size; (void)ws_size;
    const float* in  = (const float*)d_in[0];
    const float* vth = (const float*)d_in[1];  // VTH scalar (1 element)
    unsigned* ws     = (unsigned*)d_ws;
    float* out       = (float*)d_out;

    pe_init_ws<<<dim3(1), dim3(1), 0, stream>>>(ws);
    pe_minmax<<<dim3(N_V4 / 256), dim3(256), 0, stream>>>(in, ws);
    pe_encode<<<dim3(N_V4 / 256), dim3(256), 0, stream>>>(in, vth, ws, out);
}